// Steerable_2D_46377056862416
// MI455X (gfx1250) — compile-verified
//
#include <hip/hip_runtime.h>

typedef float v2f __attribute__((ext_vector_type(2)));
typedef float v8f __attribute__((ext_vector_type(8)));

#define N_V   4096
#define C0    16
#define C1    32
#define C2    32
#define NB    7
#define K1    7
#define K1S   49        // 7*7
#define K2    13
#define K2S   169       // 13*13
#define F1LEN (K1S * C1)   // 1568 floats per vertex
#define X1ROWS 64          // pad 49 -> 64
#define X2ROWS 176         // pad 169 -> 176

// ---------------------------------------------------------------------------
// Level 1: f1[v] = relu( x1 @ W1^T + b1 ),  x1 = reshape(agg1, [49,16])
// one wave per vertex, 8 waves per block
// ---------------------------------------------------------------------------
__global__ __launch_bounds__(256) void level1_kernel(
    const float* __restrict__ labels, const int* __restrict__ nbrs,
    const float* __restrict__ mask1,  const float* __restrict__ adj1,
    const float* __restrict__ W1,     const float* __restrict__ b1,
    const float* __restrict__ al1p,   float* __restrict__ f1out)
{
    __shared__ float W1s[C1 * C0];          // 512
    __shared__ float b1s[C1];
    __shared__ float agg[8][X1ROWS * C0];   // 8 * 1024 (x1, padded rows zeroed)
    __shared__ float labs[8][NB * C0];      // 8 * 112
    __shared__ float m1s[8][NB * K1];       // 8 * 49
    __shared__ float adjs[8][K1S];          // 8 * 49

    const int tid  = threadIdx.x;
    const int lane = tid & 31;
    const int w    = tid >> 5;
    const int v    = blockIdx.x * 8 + w;    // grid = N_V/8 exactly

    for (int t = tid; t < C1 * C0; t += 256) W1s[t] = W1[t];
    if (tid < C1) b1s[tid] = b1[tid];
    const float al1 = al1p[0];

    // per-wave neighborhood data
    for (int t = lane; t < NB * C0; t += 32) {
        int u = t >> 4;                // t / 16
        int c = t & 15;
        int nu = nbrs[v * NB + u];
        labs[w][t] = labels[nu * C0 + c];
    }
    for (int t = lane; t < K1S; t += 32) {
        m1s[w][t]  = mask1[v * K1S + t];
        adjs[w][t] = adj1[v * K1S + t];
    }
    __syncthreads();

    // build x1 (linear layout of agg1 [16][7][7] == reshape [49][16])
    for (int f = lane; f < X1ROWS * C0; f += 32) {
        float val = 0.f;
        if (f < C0 * K1S) {
            int c  = f / K1S;
            int kk = f - c * K1S;
            int k1 = kk / K1, k2 = kk - k1 * K1;
            val = al1 * adjs[w][kk];
            #pragma unroll
            for (int u = 0; u < NB; ++u) {
                float mp = m1s[w][u * K1 + k1] * m1s[w][u * K1 + k2];
                val += mp * labs[w][u * C0 + c];
            }
        }
        agg[w][f] = val;
    }
    __syncthreads();

    // WMMA GEMM: [64,16] x [16,32] using V_WMMA_F32_16X16X4_F32
    const int col16 = lane & 15;
    const int koff  = (lane >= 16) ? 2 : 0;
    for (int mt = 0; mt < 4; ++mt) {
        for (int nt = 0; nt < 2; ++nt) {
            v8f acc;
            float bias = b1s[nt * 16 + col16];
            #pragma unroll
            for (int i = 0; i < 8; ++i) acc[i] = bias;
            const int arow = mt * 16 + col16;
            const int bcol = nt * 16 + col16;
            #pragma unroll
            for (int kt = 0; kt < 4; ++kt) {
                v2f a, b;
                a.x = agg[w][arow * C0 + 4 * kt + koff];
                a.y = agg[w][arow * C0 + 4 * kt + koff + 1];
                b.x = W1s[bcol * C0 + 4 * kt + koff];       // B[k][n] = W1[n][k]
                b.y = W1s[bcol * C0 + 4 * kt + koff + 1];
                acc = __builtin_amdgcn_wmma_f32_16x16x4_f32(
                          false, a, false, b, (short)0, acc, false, false);
            }
            const int colg  = nt * 16 + col16;
            const int rbase = mt * 16 + ((lane >= 16) ? 8 : 0);
            #pragma unroll
            for (int i = 0; i < 8; ++i) {
                int row = rbase + i;
                if (row < K1S)
                    f1out[(size_t)v * F1LEN + row * C1 + colg] = fmaxf(acc[i], 0.f);
            }
        }
    }
}

// ---------------------------------------------------------------------------
// Level 2: one block (8 waves) per vertex
// ---------------------------------------------------------------------------
__global__ __launch_bounds__(256) void level2_kernel(
    const int* __restrict__ nbrs,  const int* __restrict__ idx2,
    const float* __restrict__ mask2, const float* __restrict__ adj2,
    const float* __restrict__ W2,  const float* __restrict__ b2,
    const float* __restrict__ al2p, const float* __restrict__ f1in,
    float* __restrict__ partials)
{
    __shared__ float x2[X2ROWS * C1];   // 5632
    __shared__ float y2[X2ROWS * C1];   // 5632
    __shared__ float W2s[C2 * C1];      // 1024
    __shared__ float b2s[C2];
    __shared__ float m2s[NB * K2];      // 91
    __shared__ int   i2s[NB * K2];      // 91
    __shared__ float adjs[K2S];         // 169
    __shared__ int   nbs[NB];

    const int tid  = threadIdx.x;
    const int lane = tid & 31;
    const int w    = tid >> 5;
    const int v    = blockIdx.x;

    for (int t = tid; t < C2 * C1; t += 256) W2s[t] = W2[t];
    if (tid < C2) b2s[tid] = b2[tid];
    for (int t = tid; t < NB * K2; t += 256) {
        m2s[t] = mask2[v * NB * K2 + t];
        i2s[t] = idx2[v * NB * K2 + t];
    }
    for (int t = tid; t < K2S; t += 256) adjs[t] = adj2[v * K2S + t];
    if (tid < NB) nbs[tid] = nbrs[v * NB + tid];
    const float al2 = al2p[0];
    __syncthreads();

    // gather-build x2 (linear layout of agg2 [32][13][13] == reshape [169][32])
    for (int f = tid; f < X2ROWS * C1; f += 256) {
        float val = 0.f;
        if (f < C1 * K2S) {
            int c = f / K2S;
            int s = f - c * K2S;
            int i = s / K2, j = s - i * K2;
            val = al2 * adjs[s];
            #pragma unroll
            for (int u = 0; u < NB; ++u) {
                float mp = m2s[u * K2 + i] * m2s[u * K2 + j];
                if (mp != 0.f) {
                    int a = i2s[u * K2 + i];
                    int b = i2s[u * K2 + j];
                    // f1[nbr][c][a][b] = f1flat[c*49 + a*7 + b] (view of [49][32] memory)
                    val += mp * f1in[(size_t)nbs[u] * F1LEN + c * K1S + a * K1 + b];
                }
            }
        }
        x2[f] = val;
    }
    __syncthreads();

    // WMMA GEMM: [176,32] x [32,32]; 22 tiles over 8 waves, K=32 (8 steps of 4)
    const int col16 = lane & 15;
    const int koff  = (lane >= 16) ? 2 : 0;
    for (int t = w; t < 22; t += 8) {
        const int mt = t >> 1, nt = t & 1;
        v8f acc;
        float bias = b2s[nt * 16 + col16];
        #pragma unroll
        for (int i = 0; i < 8; ++i) acc[i] = bias;
        const int arow = mt * 16 + col16;
        const int bcol = nt * 16 + col16;
        #pragma unroll
        for (int kt = 0; kt < 8; ++kt) {
            v2f a, b;
            a.x = x2[arow * C1 + 4 * kt + koff];
            a.y = x2[arow * C1 + 4 * kt + koff + 1];
            b.x = W2s[bcol * C1 + 4 * kt + koff];           // B[k][n] = W2[n][k]
            b.y = W2s[bcol * C1 + 4 * kt + koff + 1];
            acc = __builtin_amdgcn_wmma_f32_16x16x4_f32(
                      false, a, false, b, (short)0, acc, false, false);
        }
        const int rbase = mt * 16 + ((lane >= 16) ? 8 : 0);
        #pragma unroll
        for (int i = 0; i < 8; ++i) {
            int row = rbase + i;
            y2[row * C1 + nt * 16 + col16] = (row < K2S) ? fmaxf(acc[i], 0.f) : 0.f;
        }
    }
    __syncthreads();

    // per-channel spatial sums: channel c sums linear y2 indices [c*169, (c+1)*169)
    if (tid < C2) {
        float s = 0.f;
        for (int q = 0; q < K2S; ++q) s += y2[tid * K2S + q];
        partials[(size_t)v * C2 + tid] = s;
    }
}

// ---------------------------------------------------------------------------
// Finalize: deterministic reduction over vertices + final dot
// out[0] = g_repr . Wfc + bfc ; out[1..32] = g_repr
// ---------------------------------------------------------------------------
__global__ void finalize_kernel(const float* __restrict__ partials,
                                const float* __restrict__ Wfc,
                                const float* __restrict__ bfc,
                                float* __restrict__ out)
{
    __shared__ float g[C2];
    const int tid = threadIdx.x;
    if (tid < C2) {
        float s = 0.f;
        for (int vv = 0; vv < N_V; ++vv) s += partials[(size_t)vv * C2 + tid];
        g[tid] = s;
        out[1 + tid] = s;
    }
    __syncthreads();
    if (tid == 0) {
        float o = bfc[0];
        for (int c = 0; c < C2; ++c) o += g[c] * Wfc[c];
        out[0] = o;
    }
}

extern "C" void kernel_launch(void* const* d_in, const int* in_sizes, int n_in,
                              void* d_out, int out_size, void* d_ws, size_t ws_size,
                              hipStream_t stream)
{
    const float* labels = (const float*)d_in[0];
    const int*   nbrs   = (const int*)  d_in[1];
    /* idx1 = d_in[2] unused: Kp==1 so all gather indices are 0 */
    const float* mask1  = (const float*)d_in[3];
    const int*   idx2   = (const int*)  d_in[4];
    const float* mask2  = (const float*)d_in[5];
    const float* adj1   = (const float*)d_in[6];
    const float* adj2   = (const float*)d_in[7];
    const float* W1     = (const float*)d_in[8];
    const float* b1     = (const float*)d_in[9];
    const float* W2     = (const float*)d_in[10];
    const float* b2     = (const float*)d_in[11];
    const float* al1    = (const float*)d_in[12];
    const float* al2    = (const float*)d_in[13];
    const float* Wfc    = (const float*)d_in[14];
    const float* bfc    = (const float*)d_in[15];

    float* f1ws     = (float*)d_ws;                       // 4096*1568 floats = 25.7 MB
    float* partials = f1ws + (size_t)N_V * F1LEN;         // 4096*32 floats = 512 KB
    float* out      = (float*)d_out;                      // 33 floats

    level1_kernel<<<N_V / 8, 256, 0, stream>>>(labels, nbrs, mask1, adj1,
                                               W1, b1, al1, f1ws);
    level2_kernel<<<N_V, 256, 0, stream>>>(nbrs, idx2, mask2, adj2,
                                           W2, b2, al2, f1ws, partials);
    finalize_kernel<<<1, 32, 0, stream>>>(partials, Wfc, bfc, out);
}